// SATInstanceEncoderHetero_softmax_43087111913767
// MI455X (gfx1250) — compile-verified
//
#include <hip/hip_runtime.h>

#define N_POSL 50000
#define N_NEGL 50000
#define N_CLSC 200000
#define NEDGE  300000
#define HID    128
#define IN_D   10

typedef float v2f __attribute__((ext_vector_type(2)));
typedef float v8f __attribute__((ext_vector_type(8)));

// ---------------- workspace layout (float units) ----------------
constexpr size_t OFF_CNT_POS = 0;
constexpr size_t OFF_CNT_NEG = OFF_CNT_POS + N_POSL;
constexpr size_t OFF_CNT_CP  = OFF_CNT_NEG + N_NEGL;
constexpr size_t OFF_CNT_CN  = OFF_CNT_CP + N_CLSC;
constexpr size_t OFF_AGGP    = OFF_CNT_CN + N_CLSC;
constexpr size_t OFF_AGGN    = OFF_AGGP + (size_t)N_POSL * IN_D;
constexpr size_t OFF_AGGCP   = OFF_AGGN + (size_t)N_NEGL * IN_D;
constexpr size_t OFF_AGGCN   = OFF_AGGCP + (size_t)N_CLSC * 256;
constexpr size_t OFF_HG      = OFF_AGGCN + (size_t)N_CLSC * 256;
constexpr size_t OFF_ZEND    = OFF_HG + (size_t)N_CLSC * 2;   // zero region end
constexpr size_t OFF_HP0     = OFF_ZEND;
constexpr size_t OFF_HN0     = OFF_HP0 + (size_t)N_POSL * 256;

// ---------------- degree counting ----------------
__global__ void count_deg_kernel(const int* __restrict__ eps, const int* __restrict__ epd,
                                 const int* __restrict__ ens, const int* __restrict__ e_nd,
                                 int* __restrict__ cnt_pos, int* __restrict__ cnt_cp,
                                 int* __restrict__ cnt_neg, int* __restrict__ cnt_cn) {
  int t = blockIdx.x * blockDim.x + threadIdx.x;
  if (t >= NEDGE) return;
  atomicAdd(&cnt_pos[eps[t]], 1);
  atomicAdd(&cnt_cp[epd[t]], 1);
  atomicAdd(&cnt_neg[ens[t]], 1);
  atomicAdd(&cnt_cn[e_nd[t]], 1);
}

// ---------------- layer-0 edge scatter: clause(10) -> literal agg ----------------
__global__ void agg0_kernel(const int* __restrict__ eps, const int* __restrict__ epd,
                            const int* __restrict__ ens, const int* __restrict__ e_nd,
                            const int* __restrict__ cnt_cp, const int* __restrict__ cnt_cn,
                            const float* __restrict__ h_cls,
                            float* __restrict__ aggP, float* __restrict__ aggN) {
  int t = blockIdx.x * blockDim.x + threadIdx.x;
  if (t >= 2 * NEDGE) return;
  bool pos = t < NEDGE;
  int e   = pos ? t : t - NEDGE;
  int c   = pos ? epd[e] : e_nd[e];
  int lit = pos ? eps[e] : ens[e];
  int dc  = pos ? cnt_cp[c] : cnt_cn[c];
  float s = rsqrtf((float)(dc < 1 ? 1 : dc));
  const float* hc = h_cls + (size_t)c * IN_D;
  float* dst = (pos ? aggP : aggN) + (size_t)lit * IN_D;
#pragma unroll
  for (int d = 0; d < IN_D; ++d) atomicAdd(&dst[d], hc[d] * s);
}

// ---------------- layer-0 dense: hp0 / hn0 = relu( (agg*nd)@Wct+bct , flip@Wfl+bfl )
__global__ void lit_dense_kernel(const float* __restrict__ aggP, const float* __restrict__ aggN,
                                 const int* __restrict__ cnt_pos, const int* __restrict__ cnt_neg,
                                 const float* __restrict__ h_pos, const float* __restrict__ h_neg,
                                 const float* __restrict__ Wct, const float* __restrict__ bct,
                                 const float* __restrict__ Wfl, const float* __restrict__ bfl,
                                 float* __restrict__ hp0, float* __restrict__ hn0) {
  __shared__ float sA[IN_D], sF[IN_D];
  int p   = blockIdx.x;
  int neg = blockIdx.y;
  const float* agg   = neg ? aggN : aggP;
  const int*   cnt   = neg ? cnt_neg : cnt_pos;
  const float* hflip = neg ? h_pos : h_neg;   // pos literals pair with h_neg, neg with h_pos
  float* out = neg ? hn0 : hp0;
  int tx = threadIdx.x;
  if (tx < IN_D) {
    int d = cnt[p];
    float nd = rsqrtf((float)(d < 1 ? 1 : d));
    sA[tx] = agg[(size_t)p * IN_D + tx] * nd;
    sF[tx] = hflip[(size_t)p * IN_D + tx];
  }
  __syncthreads();
  float a0 = bct[tx], a1 = bfl[tx];
#pragma unroll
  for (int k = 0; k < IN_D; ++k) {
    a0 += sA[k] * Wct[k * HID + tx];
    a1 += sF[k] * Wfl[k * HID + tx];
  }
  out[(size_t)p * 256 + tx]       = fmaxf(a0, 0.0f);
  out[(size_t)p * 256 + HID + tx] = fmaxf(a1, 0.0f);
}

// ---------------- layer-1 edge scatter: literal(256) -> clause agg, 1 wave/edge ----
__global__ void agg1_kernel(const int* __restrict__ eps, const int* __restrict__ epd,
                            const int* __restrict__ ens, const int* __restrict__ e_nd,
                            const int* __restrict__ cnt_pos, const int* __restrict__ cnt_neg,
                            const float* __restrict__ hp0, const float* __restrict__ hn0,
                            float* __restrict__ aggCp, float* __restrict__ aggCn) {
  int wv = threadIdx.x >> 5, lane = threadIdx.x & 31;
  int id = blockIdx.x * 8 + wv;              // 0 .. 2*NEDGE-1 (grid sized exactly)
  bool pos = id < NEDGE;
  int e = pos ? id : id - NEDGE;
  int s = pos ? eps[e] : ens[e];
  int d = pos ? epd[e] : e_nd[e];
  int dg = pos ? cnt_pos[s] : cnt_neg[s];
  float ns = rsqrtf((float)(dg < 1 ? 1 : dg));
  const float* src = (pos ? hp0 : hn0) + (size_t)s * 256 + lane * 8;
  float* dst = (pos ? aggCp : aggCn) + (size_t)d * 256 + lane * 8;
  float4 v0 = *(const float4*)src;
  float4 v1 = *(const float4*)(src + 4);
  atomicAdd(dst + 0, v0.x * ns); atomicAdd(dst + 1, v0.y * ns);
  atomicAdd(dst + 2, v0.z * ns); atomicAdd(dst + 3, v0.w * ns);
  atomicAdd(dst + 4, v1.x * ns); atomicAdd(dst + 5, v1.y * ns);
  atomicAdd(dst + 6, v1.z * ns); atomicAdd(dst + 7, v1.w * ns);
}

// ---------------- layer-1 GEMM (fp32 WMMA) fused with output head ----------------
// Per relation r: M = N_CLS*2 rows (row g = c*2+j), K=N=128.
// D = relu(nd * (aggRaw @ W1_in) + b1_in); hg[c] += sum_n D * fc_W[r*256+j*128+n].
// nd (per-row scalar) is folded into the epilogue so A staging is a pure copy,
// done with gfx1250 async global->LDS (ASYNCcnt) instead of a VGPR round-trip.
__launch_bounds__(512)
__global__ void wmma_head_kernel(const float* __restrict__ aggCp, const float* __restrict__ aggCn,
                                 const int* __restrict__ cnt_cp, const int* __restrict__ cnt_cn,
                                 const float* __restrict__ W1, const float* __restrict__ b1,
                                 const float* __restrict__ fcW, float* __restrict__ hg) {
  __shared__ __align__(16) float sA[32 * 132];   // 32 rows, stride 132 (528B, 16B-aligned)
  int rel = blockIdx.y;
  const float* agg = rel ? aggCn : aggCp;
  const int*   cnt = rel ? cnt_cn : cnt_cp;
  int t = threadIdx.x;

  // Async-stage 64 contiguous A rows from the block's base (8192 floats = 32KB); sA holds 32 of them.
  {
    const float* tileBase = agg + (size_t)blockIdx.x * 32 * 128;
#pragma unroll
    for (int i = 0; i < 4; ++i) {
      int q   = t + 512 * i;                       // 16-byte chunk id, 0..2047
      int r   = q >> 5;                            // row 0..63 (32 chunks/row)
      int col = (q & 31) * 4;                      // float column
      unsigned ldsOff = (unsigned)(uintptr_t)(&sA[r * 132 + col]);
      unsigned gOff   = (unsigned)(q * 16);        // byte offset within tile
      asm volatile("global_load_async_to_lds_b128 %0, %1, %2"
                   :: "v"(ldsOff), "v"(gOff), "s"(tileBase) : "memory");
    }
    asm volatile("s_wait_asynccnt 0x0" ::: "memory");
  }
  __syncthreads();

  int wv = t >> 5, lane = t & 31;
  int mSub = wv >> 3, nTile = wv & 7;              // 2 x 8 grid of 16x16 tiles
  int l15 = lane & 15, kh = lane >> 4;
  int nn = nTile * 16 + l15;

  float bias  = b1[nn];
  float fcw00 = fcW[(rel * 256 + nn) * 2];         // j=0, out 0
  float fcw01 = fcW[(rel * 256 + nn) * 2 + 1];     // j=0, out 1
  float fcw10 = fcW[(rel * 256 + 128 + nn) * 2];   // j=1, out 0
  float fcw11 = fcW[(rel * 256 + 128 + nn) * 2 + 1];

  v8f acc = {0.f, 0.f, 0.f, 0.f, 0.f, 0.f, 0.f, 0.f};
  const float* sArow = &sA[(mSub * 16 + l15) * 132];
#pragma unroll
  for (int k0 = 0; k0 < 128; k0 += 4) {
    int kb = k0 + 2 * kh;                          // lane's K-pair base (A & B layouts agree)
    v2f a = *(const v2f*)(sArow + kb);             // A[m][kb], A[m][kb+1]
    v2f b;
    b.x = W1[kb * HID + nn];                       // B[kb][n]
    b.y = W1[(kb + 1) * HID + nn];                 // B[kb+1][n]
    acc = __builtin_amdgcn_wmma_f32_16x16x4_f32(false, a, false, b, (short)0, acc,
                                                false, false);
  }

  // Epilogue: apply nd (per-clause), bias, ReLU; fold into fc_W; 16-lane XOR
  // butterfly via ds_swizzle (and=0x1F, xor=m -> imm (m<<10)|0x1F); accumulate logits.
#pragma unroll
  for (int cc = 0; cc < 4; ++cc) {
    int cG = blockIdx.x * 16 + mSub * 8 + kh * 4 + cc;
    int dgc = cnt[cG];
    float nd = rsqrtf((float)(dgc < 1 ? 1 : dgc));
    float r0 = fmaxf(fmaf(acc[2 * cc],     nd, bias), 0.0f);   // row j = 0
    float r1 = fmaxf(fmaf(acc[2 * cc + 1], nd, bias), 0.0f);   // row j = 1
    float p0 = r0 * fcw00 + r1 * fcw10;
    float p1 = r0 * fcw01 + r1 * fcw11;
    p0 += __int_as_float(__builtin_amdgcn_ds_swizzle(__float_as_int(p0), 0x041F));
    p1 += __int_as_float(__builtin_amdgcn_ds_swizzle(__float_as_int(p1), 0x041F));
    p0 += __int_as_float(__builtin_amdgcn_ds_swizzle(__float_as_int(p0), 0x081F));
    p1 += __int_as_float(__builtin_amdgcn_ds_swizzle(__float_as_int(p1), 0x081F));
    p0 += __int_as_float(__builtin_amdgcn_ds_swizzle(__float_as_int(p0), 0x101F));
    p1 += __int_as_float(__builtin_amdgcn_ds_swizzle(__float_as_int(p1), 0x101F));
    p0 += __int_as_float(__builtin_amdgcn_ds_swizzle(__float_as_int(p0), 0x201F));
    p1 += __int_as_float(__builtin_amdgcn_ds_swizzle(__float_as_int(p1), 0x201F));
    if (l15 == 0) {
      atomicAdd(&hg[(size_t)cG * 2], p0);
      atomicAdd(&hg[(size_t)cG * 2 + 1], p1);
    }
  }
}

// ---------------- final bias + softmax ----------------
__global__ void softmax_kernel(const float* __restrict__ hg, const float* __restrict__ fcb,
                               float* __restrict__ out) {
  int t = blockIdx.x * blockDim.x + threadIdx.x;
  if (t >= N_CLSC) return;
  float a = hg[2 * t] + fcb[0];
  float b = hg[2 * t + 1] + fcb[1];
  float m = fmaxf(a, b);
  float e0 = __expf(a - m), e1 = __expf(b - m);
  float inv = 1.0f / (e0 + e1);
  out[2 * t]     = e0 * inv;
  out[2 * t + 1] = e1 * inv;
}

extern "C" void kernel_launch(void* const* d_in, const int* in_sizes, int n_in,
                              void* d_out, int out_size, void* d_ws, size_t ws_size,
                              hipStream_t stream) {
  const float* h_pos = (const float*)d_in[0];
  const float* h_neg = (const float*)d_in[1];
  const float* h_cls = (const float*)d_in[2];
  const int* eps  = (const int*)d_in[3];
  const int* epd  = (const int*)d_in[4];
  const int* ens  = (const int*)d_in[5];
  const int* e_nd = (const int*)d_in[6];
  const float* W0_ct = (const float*)d_in[9];
  const float* b0_ct = (const float*)d_in[10];
  const float* W0_fl = (const float*)d_in[11];
  const float* b0_fl = (const float*)d_in[12];
  const float* W1_in = (const float*)d_in[13];
  const float* b1_in = (const float*)d_in[14];
  const float* fcW   = (const float*)d_in[19];
  const float* fcb   = (const float*)d_in[20];
  float* out = (float*)d_out;

  float* ws = (float*)d_ws;
  int* cnt_pos = (int*)(ws + OFF_CNT_POS);
  int* cnt_neg = (int*)(ws + OFF_CNT_NEG);
  int* cnt_cp  = (int*)(ws + OFF_CNT_CP);
  int* cnt_cn  = (int*)(ws + OFF_CNT_CN);
  float* aggP  = ws + OFF_AGGP;
  float* aggN  = ws + OFF_AGGN;
  float* aggCp = ws + OFF_AGGCP;
  float* aggCn = ws + OFF_AGGCN;
  float* hg    = ws + OFF_HG;
  float* hp0   = ws + OFF_HP0;
  float* hn0   = ws + OFF_HN0;

  // Zero counts + all accumulators + logits in one capture-safe memset.
  hipMemsetAsync(d_ws, 0, OFF_ZEND * sizeof(float), stream);

  count_deg_kernel<<<(NEDGE + 255) / 256, 256, 0, stream>>>(
      eps, epd, ens, e_nd, cnt_pos, cnt_cp, cnt_neg, cnt_cn);

  agg0_kernel<<<(2 * NEDGE + 255) / 256, 256, 0, stream>>>(
      eps, epd, ens, e_nd, cnt_cp, cnt_cn, h_cls, aggP, aggN);

  lit_dense_kernel<<<dim3(N_POSL, 2), 128, 0, stream>>>(
      aggP, aggN, cnt_pos, cnt_neg, h_pos, h_neg,
      W0_ct, b0_ct, W0_fl, b0_fl, hp0, hn0);

  agg1_kernel<<<(2 * NEDGE) / 8, 256, 0, stream>>>(
      eps, epd, ens, e_nd, cnt_pos, cnt_neg, hp0, hn0, aggCp, aggCn);

  wmma_head_kernel<<<dim3((N_CLSC * 2) / 32, 2), 512, 0, stream>>>(
      aggCp, aggCn, cnt_cp, cnt_cn, W1_in, b1_in, fcW, hg);

  softmax_kernel<<<(N_CLSC + 255) / 256, 256, 0, stream>>>(hg, fcb, out);
}